// Sinusoidal_net_68736656605351
// MI455X (gfx1250) — compile-verified
//
#include <hip/hip_runtime.h>
#include <cstdint>

// HIP's float4 is a class type; use a raw ext-vector so
// __builtin_nontemporal_{load,store} accept it (lowers to *_b128 th:NT).
typedef float v4f __attribute__((ext_vector_type(4)));

// sin(pi * u) for u >= 0 via v_sin_f32 (hardware computes sin(2*pi*x)).
// Exact range reduction: r = u - 2*rint(u/2) in [-1,1]; both operands are
// exactly representable and close in magnitude, so the fma is exact.
__device__ __forceinline__ float sin_pi_u(float u) {
  float k = __builtin_rintf(u * 0.5f);
  float r = __builtin_fmaf(-2.0f, k, u);
  return __builtin_amdgcn_sinf(0.5f * r); // v_sin_f32
}

__global__ void __launch_bounds__(256)
sinusoidal_net_kernel(const float* __restrict__ t,
                      const float* __restrict__ weights,
                      const float* __restrict__ c,
                      float* __restrict__ out,
                      int n, unsigned nc) {
  extern __shared__ float lds_w[]; // nc floats (16 KB for nc=4096)
  const int tid = threadIdx.x;

  // ---- Stage weight table into LDS with CDNA5 async global->LDS copies ----
  // GVS addressing: mem = SGPR_base + VGPR_offset; VDST VGPR holds the LDS
  // byte address. Tracked by ASYNCcnt.
  {
    const unsigned lds_base =
        (unsigned)(uintptr_t)&lds_w[0]; // low 32 bits of AS(3) addr = LDS offset
    const unsigned long long wbase = (unsigned long long)(uintptr_t)weights;
    for (unsigned k4 = (unsigned)tid; k4 * 4u < nc; k4 += 256u) {
      unsigned byteoff = k4 * 16u;
      unsigned lds_addr = lds_base + byteoff;
      asm volatile("global_load_async_to_lds_b128 %0, %1, %2"
                   :: "v"(lds_addr), "v"(byteoff), "s"(wbase)
                   : "memory");
    }
    asm volatile("s_wait_asynccnt 0" ::: "memory");
    __syncthreads();
  }

  const float invc = 1.0f / c[0];
  const bool pow2 = (nc & (nc - 1u)) == 0u; // uniform branch
  const unsigned mask = nc - 1u;

  const int gid = blockIdx.x * 256 + tid;
  const int stride = gridDim.x * 256;
  const int n4 = n >> 2;
  const v4f* __restrict__ t4 = (const v4f*)t;
  v4f* __restrict__ o4 = (v4f*)out;

  // ---- Streaming main loop: b128 NT load, 4x (sin + LDS gather), b128 NT store
  for (int i = gid; i < n4; i += stride) {
    v4f tv = __builtin_nontemporal_load(&t4[i]);
    v4f ov;
#pragma unroll
    for (int j = 0; j < 4; ++j) {
      float u = tv[j] * invc;            // t/c
      float s = sin_pi_u(u);             // sin(pi * t/c)
      unsigned bin = (unsigned)(int)u;   // u >= 0: trunc == floor
      unsigned half = bin >> 1;          // full-period index
      unsigned idx = pow2 ? (half & mask) : (half % nc);
      ov[j] = lds_w[idx] * s;            // ds_load_b32 gather
    }
    __builtin_nontemporal_store(ov, &o4[i]);
  }

  // Tail (n not a multiple of 4)
  for (int i = (n4 << 2) + gid; i < n; i += stride) {
    float u = t[i] * invc;
    float s = sin_pi_u(u);
    unsigned half = ((unsigned)(int)u) >> 1;
    unsigned idx = pow2 ? (half & mask) : (half % nc);
    out[i] = lds_w[idx] * s;
  }
}

extern "C" void kernel_launch(void* const* d_in, const int* in_sizes, int n_in,
                              void* d_out, int out_size, void* d_ws, size_t ws_size,
                              hipStream_t stream) {
  const float* t = (const float*)d_in[0];       // (N,) fp32
  const float* weights = (const float*)d_in[1]; // (nc,) fp32
  const float* c = (const float*)d_in[2];       // (1,) fp32
  // d_in[3] holds num_centers, but its value == len(weights) == in_sizes[1],
  // which the host already knows without a device read (graph-capture safe).
  const int n = in_sizes[0];
  const unsigned nc = (unsigned)in_sizes[1];
  float* out = (float*)d_out;

  const int threads = 256;
  int n4 = n >> 2;
  int blocks = (n4 + threads - 1) / threads;
  if (blocks > 8192) blocks = 8192; // persistent grid-stride
  if (blocks < 1) blocks = 1;
  size_t shmem = (size_t)nc * sizeof(float);

  hipLaunchKernelGGL(sinusoidal_net_kernel, dim3(blocks), dim3(threads),
                     shmem, stream, t, weights, c, out, n, nc);
}